// LogArray_9199819948569
// MI455X (gfx1250) — compile-verified
//
#include <hip/hip_runtime.h>
#include <hip/hip_bf16.h>
#include <math.h>

// Problem constants (from reference): log_A [N,K], log_B [K,M], out [N,M]
#define N_DIM 1024
#define K_DIM 512
#define M_DIM 1024

typedef __attribute__((ext_vector_type(2))) float v2f;
typedef __attribute__((ext_vector_type(8))) float v8f;

#define LN2F 0.6931471805599453f

// Fast log: raw v_log_f32 (log2) * ln2.  Accumulators are sums of 512
// positive terms with max term within a few e-folds of 1 -> no denorm risk.
__device__ __forceinline__ float fast_log(float x) {
    return __builtin_amdgcn_logf(x) * LN2F;
}

// ---------------------------------------------------------------------------
// Kernel 1a: row max of A + P = exp(A - rowmax), fused.
// One wave32 per row: lane keeps its 16 strided values in registers,
// shfl_xor max-reduce (result in every lane), then exp-store.
// grid = N/8 blocks of 256 threads (8 waves).
// ---------------------------------------------------------------------------
__global__ void rowmax_exp_kernel(const float* __restrict__ A,
                                  float* __restrict__ rowmax,
                                  float* __restrict__ P) {
    const int lane = threadIdx.x & 31;
    const int wave = threadIdx.x >> 5;
    const int row  = blockIdx.x * 8 + wave;
    const float* a = A + row * K_DIM;

    float v[K_DIM / 32];
    float mx = -INFINITY;
    #pragma unroll
    for (int i = 0; i < K_DIM / 32; ++i) {
        v[i] = a[lane + 32 * i];
        mx = fmaxf(mx, v[i]);
    }
    #pragma unroll
    for (int off = 16; off > 0; off >>= 1) {
        mx = fmaxf(mx, __shfl_xor(mx, off, 32));
    }
    float* p = P + row * K_DIM;
    #pragma unroll
    for (int i = 0; i < K_DIM / 32; ++i) {
        p[lane + 32 * i] = __expf(v[i] - mx);
    }
    if (lane == 0) rowmax[row] = mx;
}

// ---------------------------------------------------------------------------
// Kernel 1b: col max of B + Q = exp(B - colmax), fused (two coalesced passes;
// second pass hits L2).  One thread per column.  grid = M/256 x 256.
// ---------------------------------------------------------------------------
__global__ void colmax_exp_kernel(const float* __restrict__ B,
                                  float* __restrict__ colmax,
                                  float* __restrict__ Q) {
    const int m = blockIdx.x * blockDim.x + threadIdx.x;
    float mx = -INFINITY;
    #pragma unroll 8
    for (int k = 0; k < K_DIM; ++k) {
        mx = fmaxf(mx, B[k * M_DIM + m]);
    }
    colmax[m] = mx;
    #pragma unroll 8
    for (int k = 0; k < K_DIM; ++k) {
        Q[k * M_DIM + m] = __expf(B[k * M_DIM + m] - mx);
    }
}

// ---------------------------------------------------------------------------
// Kernel 2: pure exp-space GEMM from precomputed P,Q via V_WMMA_F32_16X16X4.
//
// Block = 256 threads = 8 waves laid out 4(row) x 2(col); block tile is
// 128 rows x 64 cols.  Each wave owns a 32(M) x 32(N) strip:
// 2 row-groups x 2 col-tiles = 4 f32 accumulators.  Per k-step (K+=4):
//   2 x b64 A loads + 4 x b32 B loads -> 4 WMMAs (B fragments reused by
//   both row-groups).  grid = (M/64, N/128) = 128 blocks -> 1024 waves,
// sized for WGP coverage (the whole working set is L2-resident, so total
// L2 traffic is ~independent of tile width; occupancy dominates).
//
// Operand lane layouts (ISA 7.12.2, 32-bit):
//   A 16x4 : lanes 0-15 -> K={k0,k0+1}; lanes 16-31 -> K={k0+2,k0+3}; M=lane&15
//   B 4x16 : vgpr0 = K rows {k0 | k0+2}, vgpr1 = {k0+1 | k0+3}; N=lane&15
//   C/D    : vgpr r: rows r (lanes 0-15) / r+8 (lanes 16-31); col = lane&15
// ---------------------------------------------------------------------------
__global__ void gemm_pq_kernel(const float* __restrict__ P,
                               const float* __restrict__ Q,
                               const float* __restrict__ rowmax,
                               const float* __restrict__ colmax,
                               float* __restrict__ out) {
    const int lane  = threadIdx.x & 31;
    const int wave  = threadIdx.x >> 5;
    const int row0  = blockIdx.y * 128 + (wave & 3) * 32;
    const int col0  = blockIdx.x * 64  + (wave >> 2) * 32;

    const int m     = lane & 15;
    const int khalf = lane >> 4;

    float cm[2];
    #pragma unroll
    for (int t = 0; t < 2; ++t) cm[t] = colmax[col0 + t * 16 + m];

    v8f acc[2][2];
    #pragma unroll
    for (int g = 0; g < 2; ++g)
        #pragma unroll
        for (int t = 0; t < 2; ++t) acc[g][t] = (v8f)(0.0f);

    const float* p0 = P + (row0 + m) * K_DIM + 2 * khalf;   // rows row0..+15
    const float* p1 = p0 + 16 * K_DIM;                      // rows row0+16..+31

    for (int k0 = 0; k0 < K_DIM; k0 += 4) {
        const int ka = k0 + 2 * khalf;

        const v2f a0 = *(const v2f*)(p0 + k0);   // 8B-aligned b64 load
        const v2f a1 = *(const v2f*)(p1 + k0);

        #pragma unroll
        for (int t = 0; t < 2; ++t) {
            const int c = col0 + t * 16 + m;
            v2f b;
            b.x = Q[ka       * M_DIM + c];
            b.y = Q[(ka + 1) * M_DIM + c];

            acc[0][t] = __builtin_amdgcn_wmma_f32_16x16x4_f32(
                false, a0, false, b, (short)0, acc[0][t], false, false);
            acc[1][t] = __builtin_amdgcn_wmma_f32_16x16x4_f32(
                false, a1, false, b, (short)0, acc[1][t], false, false);
        }
    }

    // Epilogue: out = log(sum) + rowmax + colmax  (hardware v_log_f32)
    #pragma unroll
    for (int g = 0; g < 2; ++g) {
        #pragma unroll
        for (int t = 0; t < 2; ++t) {
            const int col = col0 + t * 16 + m;
            #pragma unroll
            for (int r = 0; r < 8; ++r) {
                const int row = row0 + g * 16 + r + 8 * khalf;
                out[row * M_DIM + col] =
                    fast_log(acc[g][t][r]) + rowmax[row] + cm[t];
            }
        }
    }
}

// ---------------------------------------------------------------------------
// Fallback (small workspace): fused exp-in-loop GEMM, 16x64 strip per wave.
// Needs only rowmax/colmax (8 KB) in workspace.
// ---------------------------------------------------------------------------
__global__ void rowmax_kernel(const float* __restrict__ A,
                              float* __restrict__ rowmax) {
    const int lane = threadIdx.x & 31;
    const int wave = threadIdx.x >> 5;
    const int row  = blockIdx.x * 8 + wave;
    const float* a = A + row * K_DIM;
    float mx = -INFINITY;
    #pragma unroll
    for (int k = lane; k < K_DIM; k += 32) mx = fmaxf(mx, a[k]);
    #pragma unroll
    for (int off = 16; off > 0; off >>= 1) mx = fmaxf(mx, __shfl_xor(mx, off, 32));
    if (lane == 0) rowmax[row] = mx;
}

__global__ void colmax_kernel(const float* __restrict__ B,
                              float* __restrict__ colmax) {
    const int m = blockIdx.x * blockDim.x + threadIdx.x;
    float mx = -INFINITY;
    #pragma unroll 8
    for (int k = 0; k < K_DIM; ++k) mx = fmaxf(mx, B[k * M_DIM + m]);
    colmax[m] = mx;
}

__global__ void logmm_fused_kernel(const float* __restrict__ A,
                                   const float* __restrict__ B,
                                   const float* __restrict__ rowmax,
                                   const float* __restrict__ colmax,
                                   float* __restrict__ out) {
    const int lane  = threadIdx.x & 31;
    const int wave  = threadIdx.x >> 5;
    const int row0  = blockIdx.y * 128 + wave * 16;
    const int col0  = blockIdx.x * 64;
    const int m     = lane & 15;
    const int khalf = lane >> 4;

    const float rm = rowmax[row0 + m];
    float cm[4];
    #pragma unroll
    for (int t = 0; t < 4; ++t) cm[t] = colmax[col0 + t * 16 + m];

    v8f acc[4];
    #pragma unroll
    for (int t = 0; t < 4; ++t) acc[t] = (v8f)(0.0f);

    const float* aRow = A + (row0 + m) * K_DIM;

    for (int k0 = 0; k0 < K_DIM; k0 += 4) {
        const int ka = k0 + 2 * khalf;
        v2f afrag;
        afrag.x = __expf(aRow[ka]     - rm);
        afrag.y = __expf(aRow[ka + 1] - rm);
        #pragma unroll
        for (int t = 0; t < 4; ++t) {
            const int c = col0 + t * 16 + m;
            v2f b;
            b.x = __expf(B[ka       * M_DIM + c] - cm[t]);
            b.y = __expf(B[(ka + 1) * M_DIM + c] - cm[t]);
            acc[t] = __builtin_amdgcn_wmma_f32_16x16x4_f32(
                false, afrag, false, b, (short)0, acc[t], false, false);
        }
    }

    #pragma unroll
    for (int t = 0; t < 4; ++t) {
        const int col = col0 + t * 16 + m;
        #pragma unroll
        for (int r = 0; r < 8; ++r) {
            const int row = row0 + r + 8 * khalf;
            out[row * M_DIM + col] = fast_log(acc[t][r]) + rowmax[row] + cm[t];
        }
    }
}

// ---------------------------------------------------------------------------
// Launcher.  d_in[0] = log_A, d_in[1] = log_B (f32); d_out = 1024x1024 f32.
// Preferred path workspace: rowmax(4KB) + colmax(4KB) + P(2MB) + Q(2MB).
// ---------------------------------------------------------------------------
extern "C" void kernel_launch(void* const* d_in, const int* in_sizes, int n_in,
                              void* d_out, int out_size, void* d_ws, size_t ws_size,
                              hipStream_t stream) {
    const float* A = (const float*)d_in[0];
    const float* B = (const float*)d_in[1];
    float* out     = (float*)d_out;

    float* rowmax = (float*)d_ws;
    float* colmax = rowmax + N_DIM;

    const size_t need_full =
        (size_t)(N_DIM + M_DIM) * sizeof(float) +
        (size_t)(N_DIM * K_DIM + K_DIM * M_DIM) * sizeof(float);

    if (ws_size >= need_full) {
        float* P = colmax + M_DIM;           // [N, K]
        float* Q = P + N_DIM * K_DIM;        // [K, M]

        rowmax_exp_kernel<<<N_DIM / 8, 256, 0, stream>>>(A, rowmax, P);
        colmax_exp_kernel<<<M_DIM / 256, 256, 0, stream>>>(B, colmax, Q);

        dim3 grid(M_DIM / 64, N_DIM / 128);  // (16, 8) = 128 blocks, 1024 waves
        gemm_pq_kernel<<<grid, 256, 0, stream>>>(P, Q, rowmax, colmax, out);
    } else {
        rowmax_kernel<<<N_DIM / 8, 256, 0, stream>>>(A, rowmax);
        colmax_kernel<<<M_DIM / 256, 256, 0, stream>>>(B, colmax);

        dim3 grid(M_DIM / 64, N_DIM / 128);  // (16, 8)
        logmm_fused_kernel<<<grid, 256, 0, stream>>>(A, B, rowmax, colmax, out);
    }
}